// ICMIQNDecoder_29575144801159
// MI455X (gfx1250) — compile-verified
//
#include <hip/hip_runtime.h>
#include <hip/hip_bf16.h>

typedef __attribute__((ext_vector_type(2))) float v2f;
typedef __attribute__((ext_vector_type(8))) float v8f;

#define PI_F 3.14159265358979323846f
#define HDIM 256
#define RS   260   // padded LDS row stride (floats) -> conflict-free b64 A loads
#define CS   20    // padded CE row stride
#define SPB  4     // samples per workgroup (M = 64 rows)

// ---- dynamic LDS pool layout (float offsets) ----
#define F_HP    0        // hpad       : 16*RS  = 4160  (rows 0..3 = h of 4 samples, 4..15 = 0)
#define F_CE    4160     // CE         : 64*CS  = 1280
#define F_TAU   5440     // tau        : 64
#define F_RED   5504     // reductions : 512
#define F_LAM   6016     // lambda     : 4
#define F_SLOG  6020     // s_logit    : 4
#define F_ZQ    6024     // zq         : 4*256 = 1024
#define F_U     7168     // U          : 64*RS = 16640
#define F_TE    23808    // TE         : 16640
#define F_Z     40448    // Z          : 16640   -> end 57088
#define F_QARR  7168     // q partials : 76*256 = 19456 (aliases U/TE, dead after GEMM3)
#define F_QSUM  5504     // q sums     : 76      (aliases RED, dead by then)
#define SMEM_FLOATS 57088
#define SMEM_BYTES  (SMEM_FLOATS * 4)

__device__ __forceinline__ float softplus_f(float x) {
    return (x > 20.f) ? x : log1pf(expf(x));
}

// ---------------------------------------------------------------------------
// Kernel 0: z_eval[j][n] = (te(tau_eval_j) @ wq1[257:513])[n] for the 3 fixed
// eval taus -- computed once, makes the whole q_eval branch O(B*256).
// ---------------------------------------------------------------------------
__global__ __launch_bounds__(256) void iqn_zeval(
    const float* __restrict__ wt1, const float* __restrict__ bt1,
    const float* __restrict__ wt2, const float* __restrict__ bt2,
    const float* __restrict__ wq1, float* __restrict__ zeval)
{
    __shared__ float su[3 * HDIM];
    __shared__ float ste[3 * HDIM];
    const int n = threadIdx.x;
    const float taus[3] = {0.9f, 0.95f, 0.99f};

    for (int j = 0; j < 3; ++j) {
        float s = bt1[n];
        #pragma unroll
        for (int k = 0; k < 16; ++k)
            s += cosf(PI_F * taus[j] * (float)(k + 1)) * wt1[k * HDIM + n];
        su[j * HDIM + n] = fmaxf(s, 0.f);
    }
    __syncthreads();
    for (int j = 0; j < 3; ++j) {
        float s = bt2[n];
        for (int k = 0; k < HDIM; ++k)
            s += su[j * HDIM + k] * wt2[k * HDIM + n];
        ste[j * HDIM + n] = fmaxf(s, 0.f);
    }
    __syncthreads();
    for (int j = 0; j < 3; ++j) {
        float s = 0.f;
        for (int k = 0; k < HDIM; ++k)
            s += ste[j * HDIM + k] * wq1[(257 + k) * HDIM + n];
        zeval[j * HDIM + n] = s;
    }
}

// ---------------------------------------------------------------------------
// Kernel 1: 4 samples / workgroup, 256 threads = 8 wave32.
// te-path: M = 64 rows (4 M-tiles). Every B-tile load feeds 4 WMMAs.
// zq-path: GEMM0 with the 4 h-rows zero-padded to one M=16 tile.
// ---------------------------------------------------------------------------
__global__ __launch_bounds__(256) void iqn_main(
    const float* __restrict__ hbuf, const float* __restrict__ targets,
    const float* __restrict__ tau,
    const float* __restrict__ ws,  const float* __restrict__ bs,
    const float* __restrict__ wc,  const float* __restrict__ bc,
    const float* __restrict__ wt1, const float* __restrict__ bt1,
    const float* __restrict__ wt2, const float* __restrict__ bt2,
    const float* __restrict__ wq1, const float* __restrict__ bq1,
    const float* __restrict__ wq2, const float* __restrict__ bq2,
    const float* __restrict__ zeval,
    float* __restrict__ partials, int nB)
{
    extern __shared__ float sm[];
    float* hp   = sm + F_HP;
    float* ce   = sm + F_CE;
    float* tsh  = sm + F_TAU;
    float* red  = sm + F_RED;
    float* lamL = sm + F_LAM;
    float* slog = sm + F_SLOG;
    float* zq   = sm + F_ZQ;
    float* ub   = sm + F_U;
    float* teb  = sm + F_TE;
    float* zb   = sm + F_Z;
    float* qarr = sm + F_QARR;
    float* qsum = sm + F_QSUM;

    const int b0   = blockIdx.x * SPB;
    const int nval = min(SPB, nB - b0);
    const int tid  = threadIdx.x;
    const int lane = tid & 31;
    const int wave = tid >> 5;
    const int m     = lane & 15;            // A row / C column within a tile
    const int khalf = (lane < 16) ? 0 : 2;  // A/B K sub-pair (ISA 16x4 layout)
    const int rbase = (lane < 16) ? 0 : 8;  // C/D row striping

    // ---- stage: hpad (rows 0..3 = h, 4..15 = 0), tau, CE ----
    for (int idx = tid; idx < SPB * HDIM; idx += 256) {
        const int s = idx >> 8, k = idx & 255;
        hp[s * RS + k] = (s < nval) ? hbuf[(size_t)b0 * HDIM + idx] : 0.f;
    }
    for (int idx = tid; idx < (16 - SPB) * HDIM; idx += 256)
        hp[(SPB + (idx >> 8)) * RS + (idx & 255)] = 0.f;
    if (tid < SPB * 16)
        tsh[tid] = (tid < nval * 16) ? tau[(size_t)b0 * 16 + tid] : 0.f;
    for (int idx = tid; idx < 64 * 16; idx += 256) {
        const int row = idx >> 4, k = idx & 15;     // row = s*16 + t
        const float tv = (row < nval * 16) ? tau[(size_t)b0 * 16 + row] : 0.f;
        ce[row * CS + k] = cosf(PI_F * tv * (float)(k + 1));
    }
    __syncthreads();

    // ---- per-sample dots: s_logit = h.ws + bs ; lambda = softplus(...) + 1e-6 ----
    {
        const int g = tid >> 6, i = tid & 63;
        float ps = 0.f, pc = 0.f;
        #pragma unroll
        for (int r = 0; r < 4; ++r) {
            const int k = i + r * 64;
            const float hv = hp[g * RS + k];
            ps += hv * ws[k];
            pc += hv * wc[k];
        }
        red[g * 64 + i] = ps;
        red[256 + g * 64 + i] = pc;
    }
    __syncthreads();
    for (int off = 32; off; off >>= 1) {
        const int g = tid >> 6, i = tid & 63;
        if (i < off) {
            red[g * 64 + i]       += red[g * 64 + i + off];
            red[256 + g * 64 + i] += red[256 + g * 64 + i + off];
        }
        __syncthreads();
    }
    if (tid < SPB) {
        const float sl = red[tid * 64] + bs[0];
        const float lm = softplus_f(red[256 + tid * 64] + sl * wc[HDIM] + bc[0]) + 1e-6f;
        slog[tid] = sl;
        lamL[tid] = lm;
    }
    __syncthreads();

    // ---- GEMM1: U = relu(CE(64x16) @ wt1(16x256) + bt1), 4 M-tiles / B-load ----
    for (int jj = 0; jj < 2; ++jj) {
        const int nl = (wave + jj * 8) * 16 + m;
        v8f acc[SPB];
        #pragma unroll
        for (int s = 0; s < SPB; ++s) acc[s] = (v8f){0.f,0.f,0.f,0.f,0.f,0.f,0.f,0.f};
        #pragma unroll
        for (int s = 0; s < 4; ++s) {
            const int kk = 4 * s + khalf;
            v2f bm; bm.x = wt1[kk * HDIM + nl]; bm.y = wt1[(kk + 1) * HDIM + nl];
            #pragma unroll
            for (int smt = 0; smt < SPB; ++smt) {
                v2f a = *(const v2f*)(ce + (smt * 16 + m) * CS + kk);
                acc[smt] = __builtin_amdgcn_wmma_f32_16x16x4_f32(
                    false, a, false, bm, (short)0, acc[smt], false, false);
            }
        }
        const float bb = bt1[nl];
        #pragma unroll
        for (int smt = 0; smt < SPB; ++smt)
            #pragma unroll
            for (int r = 0; r < 8; ++r)
                ub[(smt * 16 + rbase + r) * RS + nl] = fmaxf(acc[smt][r] + bb, 0.f);
    }

    // ---- GEMM0: zq-core = Hpad(16x256) @ wq1[0:256] ; rows 0..3 are samples ----
    for (int jj = 0; jj < 2; ++jj) {
        const int nl = (wave + jj * 8) * 16 + m;
        v8f acc = {0.f,0.f,0.f,0.f,0.f,0.f,0.f,0.f};
        #pragma unroll 4
        for (int s = 0; s < 64; ++s) {
            const int kk = 4 * s + khalf;
            v2f bm; bm.x = wq1[kk * HDIM + nl]; bm.y = wq1[(kk + 1) * HDIM + nl];
            v2f a = *(const v2f*)(hp + m * RS + kk);
            acc = __builtin_amdgcn_wmma_f32_16x16x4_f32(
                false, a, false, bm, (short)0, acc, false, false);
        }
        if (lane < 16) {
            const float bq = bq1[nl];
            const float wm = wq1[256 * HDIM + nl];
            #pragma unroll
            for (int r = 0; r < SPB; ++r)          // only rows 0..3 are real samples
                zq[r * HDIM + nl] = acc[r] + bq + lamL[r] * wm;
        }
    }
    __syncthreads();

    // ---- GEMM2: TE = relu(U(64x256) @ wt2(256x256) + bt2) ----
    for (int jj = 0; jj < 2; ++jj) {
        const int nl = (wave + jj * 8) * 16 + m;
        v8f acc[SPB];
        #pragma unroll
        for (int s = 0; s < SPB; ++s) acc[s] = (v8f){0.f,0.f,0.f,0.f,0.f,0.f,0.f,0.f};
        #pragma unroll 4
        for (int s = 0; s < 64; ++s) {
            const int kk = 4 * s + khalf;
            v2f bm; bm.x = wt2[kk * HDIM + nl]; bm.y = wt2[(kk + 1) * HDIM + nl];
            #pragma unroll
            for (int smt = 0; smt < SPB; ++smt) {
                v2f a = *(const v2f*)(ub + (smt * 16 + m) * RS + kk);
                acc[smt] = __builtin_amdgcn_wmma_f32_16x16x4_f32(
                    false, a, false, bm, (short)0, acc[smt], false, false);
            }
        }
        const float bb = bt2[nl];
        #pragma unroll
        for (int smt = 0; smt < SPB; ++smt)
            #pragma unroll
            for (int r = 0; r < 8; ++r)
                teb[(smt * 16 + rbase + r) * RS + nl] = fmaxf(acc[smt][r] + bb, 0.f);
    }
    __syncthreads();

    // ---- GEMM3: Z = TE(64x256) @ wq1[257:513] ----
    for (int jj = 0; jj < 2; ++jj) {
        const int nl = (wave + jj * 8) * 16 + m;
        v8f acc[SPB];
        #pragma unroll
        for (int s = 0; s < SPB; ++s) acc[s] = (v8f){0.f,0.f,0.f,0.f,0.f,0.f,0.f,0.f};
        #pragma unroll 4
        for (int s = 0; s < 64; ++s) {
            const int kk = 4 * s + khalf;
            v2f bm; bm.x = wq1[(257 + kk) * HDIM + nl];
                    bm.y = wq1[(258 + kk) * HDIM + nl];
            #pragma unroll
            for (int smt = 0; smt < SPB; ++smt) {
                v2f a = *(const v2f*)(teb + (smt * 16 + m) * RS + kk);
                acc[smt] = __builtin_amdgcn_wmma_f32_16x16x4_f32(
                    false, a, false, bm, (short)0, acc[smt], false, false);
            }
        }
        #pragma unroll
        for (int smt = 0; smt < SPB; ++smt)
            #pragma unroll
            for (int r = 0; r < 8; ++r)
                zb[(smt * 16 + rbase + r) * RS + nl] = acc[smt][r];
    }
    __syncthreads();   // U/TE now dead -> qarr alias becomes legal

    // ---- q[s][t] = sum_n relu(zq[s][n] + Z[s*16+t][n]) * wq2[n] + bq2 ----
    {
        const int n = tid;
        const float w  = wq2[n];
        const float z0 = zeval[n], z1 = zeval[HDIM + n], z2 = zeval[2 * HDIM + n];
        for (int s = 0; s < SPB; ++s) {
            const float zqv = zq[s * HDIM + n];
            #pragma unroll
            for (int t = 0; t < 16; ++t)
                qarr[(s * 19 + t) * HDIM + n] =
                    fmaxf(zqv + zb[(s * 16 + t) * RS + n], 0.f) * w;
            qarr[(s * 19 + 16) * HDIM + n] = fmaxf(zqv + z0, 0.f) * w;
            qarr[(s * 19 + 17) * HDIM + n] = fmaxf(zqv + z1, 0.f) * w;
            qarr[(s * 19 + 18) * HDIM + n] = fmaxf(zqv + z2, 0.f) * w;
        }
    }
    __syncthreads();
    if (tid < SPB * 19) {
        float s = 0.f;
        for (int i = 0; i < HDIM; ++i) s += qarr[tid * HDIM + i];   // fixed order
        qsum[tid] = s + bq2[0];
    }
    __syncthreads();

    // ---- per-sample losses -> per-sample partials (deterministic) ----
    if (tid < nval) {
        const int   s  = tid;
        const float tg = targets[b0 + s];
        const float y  = (tg > 0.f) ? 1.f : 0.f;
        float main_s = 0.f;
        for (int t = 0; t < 16; ++t) {
            const float q   = qsum[s * 19 + t];
            const float err = tg - q;
            const float ae  = fabsf(err);
            const float hub = (ae <= 1.f) ? 0.5f * err * err : (ae - 0.5f);  // KAPPA=1
            const float wgt = fabsf(tsh[s * 16 + t] - ((err < 0.f) ? 1.f : 0.f));
            main_s += wgt * hub;
        }
        main_s *= y;
        const float x  = slog[s];
        const float sl = fmaxf(x, 0.f) - x * y + log1pf(expf(-fabsf(x)));
        const float lg = log1pf(fmaxf(tg, 0.f));
        const float cl = (lamL[s] - lg) * (lamL[s] - lg);
        float* p = partials + (size_t)(b0 + s) * 8;
        p[0] = main_s;
        p[1] = y * qsum[s * 19 + 16];
        p[2] = y * qsum[s * 19 + 17];
        p[3] = y * qsum[s * 19 + 18];
        p[4] = sl; p[5] = cl; p[6] = y; p[7] = 0.f;
    }
}

// ---------------------------------------------------------------------------
// Kernel 2: deterministic final reduction over per-sample partials -> 7 outputs
// ---------------------------------------------------------------------------
__global__ __launch_bounds__(256) void iqn_final(
    const float* __restrict__ partials, int nb, float* __restrict__ out)
{
    __shared__ float sh[7 * 256];
    float a[7] = {0.f, 0.f, 0.f, 0.f, 0.f, 0.f, 0.f};
    for (int i = threadIdx.x; i < nb; i += 256) {
        const float* p = partials + (size_t)i * 8;
        #pragma unroll
        for (int c = 0; c < 7; ++c) a[c] += p[c];
    }
    #pragma unroll
    for (int c = 0; c < 7; ++c) sh[c * 256 + threadIdx.x] = a[c];
    __syncthreads();
    for (int off = 128; off; off >>= 1) {
        if (threadIdx.x < off)
            for (int c = 0; c < 7; ++c)
                sh[c * 256 + threadIdx.x] += sh[c * 256 + threadIdx.x + off];
        __syncthreads();
    }
    if (threadIdx.x == 0) {
        const float S_main = sh[0 * 256], Sq0 = sh[1 * 256], Sq1 = sh[2 * 256],
                    Sq2 = sh[3 * 256], Ss = sh[4 * 256], Sc = sh[5 * 256],
                    Sy = sh[6 * 256];
        const float s_loss = Ss / (float)nb;
        const float c_loss = Sc / (float)nb;
        const float n_pos  = fmaxf(Sy, 1.f);
        const float main_l = S_main / (n_pos * 16.f);
        out[0] = main_l + 0.05f * s_loss + 0.05f * c_loss;
        out[1] = main_l;
        out[2] = s_loss;
        out[3] = c_loss;
        out[4] = Sq0 / n_pos;
        out[5] = Sq1 / n_pos;
        out[6] = Sq2 / n_pos;
    }
}

extern "C" void kernel_launch(void* const* d_in, const int* in_sizes, int n_in,
                              void* d_out, int out_size, void* d_ws, size_t ws_size,
                              hipStream_t stream)
{
    const float* h   = (const float*)d_in[0];
    const float* tg  = (const float*)d_in[1];
    const float* tau = (const float*)d_in[2];
    const float* ws  = (const float*)d_in[3];
    const float* bs  = (const float*)d_in[4];
    const float* wc  = (const float*)d_in[5];
    const float* bc  = (const float*)d_in[6];
    const float* wt1 = (const float*)d_in[7];
    const float* bt1 = (const float*)d_in[8];
    const float* wt2 = (const float*)d_in[9];
    const float* bt2 = (const float*)d_in[10];
    const float* wq1 = (const float*)d_in[11];
    const float* bq1 = (const float*)d_in[12];
    const float* wq2 = (const float*)d_in[13];
    const float* bq2 = (const float*)d_in[14];

    const int B = in_sizes[0] / HDIM;

    float* zeval    = (float*)d_ws;                   // 3*256 floats
    float* partials = (float*)((char*)d_ws + 4096);   // B*8 floats

    // Allow >64KB dynamic LDS (MI455X WGP has 320KB). Harmless if already set.
    (void)hipFuncSetAttribute((const void*)iqn_main,
                              hipFuncAttributeMaxDynamicSharedMemorySize,
                              SMEM_BYTES);

    iqn_zeval<<<1, 256, 0, stream>>>(wt1, bt1, wt2, bt2, wq1, zeval);
    iqn_main<<<(B + SPB - 1) / SPB, 256, SMEM_BYTES, stream>>>(
        h, tg, tau, ws, bs, wc, bc, wt1, bt1, wt2, bt2, wq1, bq1, wq2, bq2,
        zeval, partials, B);
    iqn_final<<<1, 256, 0, stream>>>(partials, B, (float*)d_out);
}